// GeGeLayer_5007931867440
// MI455X (gfx1250) — compile-verified
//
#include <hip/hip_runtime.h>
#include <hip/hip_bf16.h>

// Shapes from the reference
#define M_TOT 64      // batch*channels rows
#define K_IN  6480    // real input length per row (multiple of 4!)
#define K_TOT 6912    // padded K == OUT_SIZE (pad region is zero => skip in GEMM)
#define N_TOT 6912    // output features
#define SORT_N 8192   // next pow2 >= 6912

typedef __attribute__((ext_vector_type(2))) float v2f;
typedef __attribute__((ext_vector_type(8))) float v8f;

// ---------------------------------------------------------------------------
// Kernel 1: score = pad(x) @ W + b  using V_WMMA_F32_16X16X4_F32 (wave32).
// Grid: 432 blocks of 128 threads (4 waves). Each block owns a 16-column
// N-strip; wave w computes M-tile w (rows 16w..16w+15). All 4 waves read the
// same W columns -> 4x reuse hits in WGP$/L2; W crosses HBM once (191 MB,
// ~8.2 us at 23.3 TB/s == the roofline floor for this GEMM).
//
// K loop runs only over the REAL input length (6480): the zero padding of x
// contributes nothing, so no guards, no branches in the inner loop.
//
// VGPR layouts per CDNA5 ISA (CDNA5 ISA):
//   A 16x4 f32 (2 VGPRs):  lanes 0-15 -> M=lane, K={0,1}; lanes 16-31 -> K={2,3}
//   B 4x16 f32 (2 VGPRs):  lanes 0-15 -> N=lane, K row {0,1}; lanes 16-31 -> K {2,3}
//   C/D 16x16 f32 (8 VGPRs): VGPR r, lanes 0-15 -> M=r, lanes 16-31 -> M=r+8
// ---------------------------------------------------------------------------
__global__ __launch_bounds__(128)
void gege_score_wmma(const float* __restrict__ x,     // (64, 6480)
                     const float* __restrict__ W,     // (6912, 6912) [i][o]
                     const float* __restrict__ b,     // (6912,)
                     float* __restrict__ score)       // (64, 6912)
{
    const int lane  = threadIdx.x & 31;
    const int wave  = threadIdx.x >> 5;      // 0..3  -> M-tile
    const int ntile = blockIdx.x;            // 0..431
    const int l15   = lane & 15;
    const int half  = lane >> 4;             // 0 or 1
    const int n     = ntile * 16 + l15;      // output column for B/C/D
    const int mA    = wave * 16 + l15;       // A-matrix row held by this lane

    const float* __restrict__ xrow = x + (size_t)mA * K_IN;
    const float* __restrict__ wcol = W + n;

    v8f acc = {};
    for (int k0 = 0; k0 < K_IN; k0 += 4) {
        const int ka = k0 + half * 2;        // this lane's K pair base (even)
        // A fragment: one aligned 8-byte load (two consecutive f32)
        const v2f a = *(const v2f*)(xrow + ka);
        // B fragment: W[ka][n], W[ka+1][n]
        const float* __restrict__ wp = wcol + (size_t)ka * N_TOT;
        v2f bb;
        bb[0] = wp[0];
        bb[1] = wp[N_TOT];
        // Prefetch W a few K-steps ahead (lowers to global_prefetch_b8)
        __builtin_prefetch(wp + (size_t)16 * N_TOT, 0, 1);

        // 8 args: (neg_a, A, neg_b, B, c_mod, C, reuse_a, reuse_b)
        acc = __builtin_amdgcn_wmma_f32_16x16x4_f32(
            false, a, false, bb, (short)0, acc, false, false);
    }

    const float bias = b[n];
    const int mbase = wave * 16 + (half ? 8 : 0);
#pragma unroll
    for (int r = 0; r < 8; ++r) {
        score[(size_t)(mbase + r) * N_TOT + n] = acc[r] + bias;
    }
}

// ---------------------------------------------------------------------------
// Kernel 2: per-row bitonic argsort in LDS, then gather + inverse permutation.
// One 1024-thread workgroup per row (64 rows). 64 KB LDS (key f32 + idx i32).
// argsort ascending; pad with +INF so pads sink past position 6911.
// out[row][j]  = xf_row0[sorted_idx[j]]  (take() quirk: always row 0 of x)
// rev[row][sorted_idx[j]] = j
// ---------------------------------------------------------------------------
__global__ __launch_bounds__(1024)
void gege_sort_gather(const float* __restrict__ score,  // (64, 6912)
                      const float* __restrict__ x,      // (64, 6480); row 0 used
                      float* __restrict__ out,          // (64, 6912)
                      float* __restrict__ rev)          // (64, 6912) reverse idx
{
    __shared__ float key[SORT_N];
    __shared__ int   idx[SORT_N];

    const int row = blockIdx.x;
    const int tid = threadIdx.x;
    const float* __restrict__ srow = score + (size_t)row * N_TOT;

    for (int i = tid; i < SORT_N; i += 1024) {
        key[i] = (i < N_TOT) ? srow[i] : __builtin_inff();
        idx[i] = i;
    }
    __syncthreads();

    // Bitonic sort, ascending
    for (int k = 2; k <= SORT_N; k <<= 1) {
        for (int j = k >> 1; j > 0; j >>= 1) {
            for (int i = tid; i < SORT_N; i += 1024) {
                const int ixj = i ^ j;
                if (ixj > i) {
                    const bool up = ((i & k) == 0);
                    const float ki = key[i];
                    const float kj = key[ixj];
                    if ((ki > kj) == up) {
                        key[i] = kj; key[ixj] = ki;
                        const int t = idx[i]; idx[i] = idx[ixj]; idx[ixj] = t;
                    }
                }
            }
            __syncthreads();
        }
    }

    // Gather from padded row 0 of x; scatter inverse permutation.
    float* __restrict__ orow = out + (size_t)row * N_TOT;
    float* __restrict__ rrow = rev + (size_t)row * N_TOT;
    for (int j = tid; j < N_TOT; j += 1024) {
        const int id = idx[j];                      // id in [0, N_TOT)
        orow[j]  = (id < K_IN) ? x[id] : 0.0f;      // xf row0, zero padding
        rrow[id] = (float)j;                        // inverse permutation
    }
}

extern "C" void kernel_launch(void* const* d_in, const int* in_sizes, int n_in,
                              void* d_out, int out_size, void* d_ws, size_t ws_size,
                              hipStream_t stream) {
    const float* x = (const float*)d_in[0];   // 64*6480
    const float* W = (const float*)d_in[1];   // 6912*6912
    const float* b = (const float*)d_in[2];   // 6912

    float* out   = (float*)d_out;                       // (64,6912) gathered values
    float* rev   = out + (size_t)M_TOT * N_TOT;         // (64,6912) reverse indices
    float* score = (float*)d_ws;                        // (64,6912) scratch, 1.77 MB

    gege_score_wmma<<<N_TOT / 16, 128, 0, stream>>>(x, W, b, score);
    gege_sort_gather<<<M_TOT, 1024, 0, stream>>>(score, x, out, rev);
}